// FilteringLayer_80942953660428
// MI455X (gfx1250) — compile-verified
//
#include <hip/hip_runtime.h>
#include <math.h>

// ---------------------------------------------------------------------------
// Ragged 1-query attention, fused single pass over theta.
//   B=256, T=2560, D0=128, 64 groups with sizes cycling 16/32/48/64.
//   out[b,g,d] = sum_{t in g} softmax_t(<s[b],theta[b,t]>/128) * theta[b,t,d]
// One 128-thread block (4 x wave32) per (b,g). Group tile staged in LDS once.
// Scores computed with V_WMMA_F32_16X16X4_F32 (f32 A/B/C -> full precision).
// ---------------------------------------------------------------------------

typedef float v2f __attribute__((ext_vector_type(2)));
typedef float v8f __attribute__((ext_vector_type(8)));

#define D0        128
#define T_TOTAL   2560
#define N_GROUPS  64
#define LDS_STRIDE 132   // 128 + 4 pad: conflict-free for both access phases

__global__ __launch_bounds__(128) void ragged_attn_kernel(
    const float* __restrict__ s_i,      // [256,128]
    const float* __restrict__ theta_i,  // [256,2560,128]
    float*       __restrict__ out)      // [256,64,128]
{
    __shared__ float lds_theta[64 * LDS_STRIDE]; // up to 64 tokens x 128 dims
    __shared__ float lds_s[D0];
    __shared__ float lds_u[64];
    __shared__ float lds_e[64];

    const int g   = blockIdx.x;   // group 0..63
    const int b   = blockIdx.y;   // batch 0..255
    const int tid = threadIdx.x;  // 0..127

    // ragged geometry: sizes cycle 16,32,48,64; cycle length 160 tokens
    const int r     = g & 3;
    const int q     = g >> 2;
    const int S     = 16 * (r + 1);                 // group size (mult of 16)
    const int start = q * 160 + 8 * r * (r + 1);    // 16*r*(r+1)/2

    const float* theta_g = theta_i + ((size_t)b * T_TOTAL + start) * D0;

    // ---- stage s[b] and the theta tile into LDS (coalesced float4) ----
    lds_s[tid] = s_i[(size_t)b * D0 + tid];

    const float4* theta_g4 = (const float4*)theta_g;
    const int n4 = S * (D0 / 4);                    // float4 elements in tile
    for (int i = tid; i < n4; i += 128) {
        const int row = i >> 5;                     // i / 32
        const int c4  = i & 31;                     // i % 32
        float4 v = theta_g4[row * (D0 / 4) + c4];
        *(float4*)&lds_theta[row * LDS_STRIDE + c4 * 4] = v;
    }
    __syncthreads();

    // ---- phase 1: scores via WMMA f32 16x16x4, one wave per 16-token tile --
    // A (16x4 f32): lanes 0-15 -> M=lane, {K0,K1}; lanes 16-31 -> M=lane-16, {K2,K3}
    // B (4x16 f32): all 16 columns = s[b] (broadcast) -> D[m,n] = u[m] in all n
    const int wave   = tid >> 5;
    const int lane   = tid & 31;
    const int ntiles = S >> 4;                      // 1..4 token-tiles
    if (wave < ntiles) {
        const int m     = lane & 15;                // matrix row within tile
        const int khalf = (lane >> 4) << 1;         // 0 (lanes<16) or 2
        const float* arow = &lds_theta[(wave * 16 + m) * LDS_STRIDE];
        v8f c = {0.f, 0.f, 0.f, 0.f, 0.f, 0.f, 0.f, 0.f};
        #pragma unroll
        for (int k0 = 0; k0 < D0; k0 += 4) {
            v2f a, bb;
            a[0]  = arow[k0 + khalf];
            a[1]  = arow[k0 + khalf + 1];
            bb[0] = lds_s[k0 + khalf];
            bb[1] = lds_s[k0 + khalf + 1];
            c = __builtin_amdgcn_wmma_f32_16x16x4_f32(
                    /*neg_a=*/false, a, /*neg_b=*/false, bb,
                    /*c_mod=*/(short)0, c, /*reuse_a=*/false, /*reuse_b=*/false);
        }
        // D column 0 lives in lane 0 (rows 0-7) and lane 16 (rows 8-15)
        if (m == 0) {
            const int base = wave * 16 + (lane >> 4) * 8;
            #pragma unroll
            for (int j = 0; j < 8; ++j)
                lds_u[base + j] = c[j] * (1.0f / 128.0f);  // divide by d0
        }
    }
    __syncthreads();

    // ---- phase 2: per-group softmax (tiny: S <= 64) ----
    float mx = -INFINITY;
    for (int t = 0; t < S; ++t) mx = fmaxf(mx, lds_u[t]);   // LDS broadcast
    if (tid < S) lds_e[tid] = __expf(lds_u[tid] - mx);
    __syncthreads();
    float denom = 0.f;
    for (int t = 0; t < S; ++t) denom += lds_e[t];
    const float inv_denom = 1.0f / denom;

    // ---- phase 3: weighted sum, thread d accumulates over tokens ----
    float acc = 0.f;
    for (int t = 0; t < S; ++t)
        acc = fmaf(lds_e[t], lds_theta[t * LDS_STRIDE + tid], acc);

    out[((size_t)b * N_GROUPS + g) * D0 + tid] = acc * inv_denom;
}

extern "C" void kernel_launch(void* const* d_in, const int* in_sizes, int n_in,
                              void* d_out, int out_size, void* d_ws, size_t ws_size,
                              hipStream_t stream) {
    (void)in_sizes; (void)n_in; (void)d_ws; (void)ws_size; (void)out_size;
    const float* s_i     = (const float*)d_in[0];   // [256,128]
    const float* theta_i = (const float*)d_in[1];   // [256,2560,128]
    float*       out     = (float*)d_out;           // [256,64,128]

    dim3 grid(N_GROUPS, 256);   // (g, b)
    dim3 block(128);
    ragged_attn_kernel<<<grid, block, 0, stream>>>(s_i, theta_i, out);
}